// decoder_24481313587866
// MI455X (gfx1250) — compile-verified
//
#include <hip/hip_runtime.h>

// CDNA5 / gfx1250 GRU decoder: gates on M, batch on N, W@h^T via
// v_wmma_f32_16x16x4_f32. Weight rows are permuted into M-slots so the
// recurrence (h_new -> next-step B operand) is lane-local.

typedef __attribute__((ext_vector_type(2))) float v2f;
typedef __attribute__((ext_vector_type(8))) float v8f;

#define HDIM 20

__device__ __forceinline__ float fast_exp(float x) {
    return __builtin_amdgcn_exp2f(x * 1.44269504088896341f);
}
__device__ __forceinline__ float fast_rcp(float x) {
    return __builtin_amdgcn_rcpf(x);
}
__device__ __forceinline__ float sigmoid_f(float x) {
    return fast_rcp(1.0f + fast_exp(-x));
}
__device__ __forceinline__ float tanh_f(float x) {
    x = fminf(fmaxf(x, -15.0f), 15.0f);
    float e = fast_exp(2.0f * x);
    return (e - 1.0f) * fast_rcp(e + 1.0f);
}

// slot index i in 0..9, lane-half g -> hidden-unit index j
__device__ __forceinline__ int jmap(int i, int g) {
    return ((i >> 1) << 2) + 2 * g + (i & 1);
}

__global__ void __launch_bounds__(256)
gru_decoder_wmma(const float* __restrict__ hidden,
                 const float* __restrict__ w_ih,
                 const float* __restrict__ w_hh,
                 const float* __restrict__ b_ih,
                 const float* __restrict__ b_hh,
                 const float* __restrict__ w_l,
                 const float* __restrict__ b_l,
                 const int*   __restrict__ step_p,
                 float* __restrict__ out,
                 int B)
{
    const int step = *step_p;
    const int lane = threadIdx.x & 31;
    const int wave = blockIdx.x * (blockDim.x >> 5) + (threadIdx.x >> 5);
    const int g    = lane >> 4;      // lane half (selects K / M sub-rows)
    const int n    = lane & 15;      // batch column within tile / A-row id
    const int nb   = wave * 16 + n;  // global batch element for this column
    const bool inb = (nb < B);

    // ---------------- Stage augmented A-matrix (W~, 64x24) into VGPRs -----
    // A layout (f32 16x16x4): lane holds row M = lane%16, elems K = 4kc+2g+v.
    // Row permutation: Mglob -> (grow, s) with s = (M>>4)*8 + (M&7),
    // grow = (M>>3)&1; gate = s/10 (r,z,n,pad); j = jmap(s%10, grow).
    float A[4][6][2];
#pragma unroll
    for (int t = 0; t < 4; ++t) {
        const int row  = 16 * t + n;
        const int grow = (row >> 3) & 1;
        const int s    = ((row >> 4) << 3) | (row & 7);
        const int gate = s / 10;               // 0=r, 1=z, 2=n, 3=pad
        const int idx  = s - gate * 10;
        const int j    = jmap(idx, grow);
        const int orig = gate * HDIM + j;      // row of original (60 x *) weights
#pragma unroll
        for (int kc = 0; kc < 6; ++kc) {
#pragma unroll
            for (int v = 0; v < 2; ++v) {
                const int K = 4 * kc + 2 * g + v;
                float w = 0.0f;
                if (gate < 3) {
                    if (K < HDIM) {
                        w = w_hh[orig * HDIM + K];
                    } else if (K == 20) {          // x0 coefficient
                        w = (gate < 2) ? w_ih[orig * 2 + 0] : 0.0f;
                    } else if (K == 21) {          // x1 coefficient
                        w = (gate < 2) ? w_ih[orig * 2 + 1] : 0.0f;
                    } else if (K == 22) {          // bias column
                        w = (gate < 2) ? (b_ih[orig] + b_hh[orig]) : b_hh[orig];
                    }                               // K==23: pad 0
                }
                A[t][kc][v] = w;
            }
        }
    }

    // ---------------- Per-lane gate constants (n-gate inn, output proj) ----
    float innW0[10], innW1[10], innB[10], wl0[10], wl1[10];
#pragma unroll
    for (int i = 0; i < 10; ++i) {
        const int j    = jmap(i, g);
        const int orig = 2 * HDIM + j;             // n-gate row
        innW0[i] = w_ih[orig * 2 + 0];
        innW1[i] = w_ih[orig * 2 + 1];
        innB[i]  = b_ih[orig];
        wl0[i]   = w_l[0 * HDIM + j];
        wl1[i]   = w_l[1 * HDIM + j];
    }
    const float bl0 = b_l[0], bl1 = b_l[1];

    // ---------------- B operand = h~^T, lane supplies K = 4kc+2g+v --------
    float Bv[6][2];
#pragma unroll
    for (int q = 0; q < 5; ++q) {
        const int j = 4 * q + 2 * g;
        Bv[q][0] = inb ? hidden[(size_t)nb * HDIM + j + 0] : 0.0f;
        Bv[q][1] = inb ? hidden[(size_t)nb * HDIM + j + 1] : 0.0f;
    }
    float xr0 = 0.0f, xr1 = 0.0f;                  // x (previous y), all lanes
    Bv[5][0] = g ? 1.0f : 0.0f;                    // g=0: x0,x1 ; g=1: 1,0
    Bv[5][1] = 0.0f;

    // ---------------- Recurrence ------------------------------------------
    for (int ts = 0; ts < step; ++ts) {
        v8f acc0 = {}, acc1 = {}, acc2 = {}, acc3 = {};
#pragma unroll
        for (int kc = 0; kc < 6; ++kc) {
            v2f b  = {Bv[kc][0], Bv[kc][1]};
            v2f a0 = {A[0][kc][0], A[0][kc][1]};
            v2f a1 = {A[1][kc][0], A[1][kc][1]};
            v2f a2 = {A[2][kc][0], A[2][kc][1]};
            v2f a3 = {A[3][kc][0], A[3][kc][1]};
            acc0 = __builtin_amdgcn_wmma_f32_16x16x4_f32(false, a0, false, b, (short)0, acc0, false, false);
            acc1 = __builtin_amdgcn_wmma_f32_16x16x4_f32(false, a1, false, b, (short)0, acc1, false, false);
            acc2 = __builtin_amdgcn_wmma_f32_16x16x4_f32(false, a2, false, b, (short)0, acc2, false, false);
            acc3 = __builtin_amdgcn_wmma_f32_16x16x4_f32(false, a3, false, b, (short)0, acc3, false, false);
        }

        // D layout: lane (g,n) holds slot s = t*8+v8 = Mglob-local gate rows.
        float sl[32];
#pragma unroll
        for (int e = 0; e < 8; ++e) {
            sl[e]      = acc0[e];
            sl[8 + e]  = acc1[e];
            sl[16 + e] = acc2[e];
            sl[24 + e] = acc3[e];
        }

        float hnew[10];
        float py0 = 0.0f, py1 = 0.0f;
#pragma unroll
        for (int i = 0; i < 10; ++i) {
            float r   = sigmoid_f(sl[i]);          // slots 0..9   : ir+hr+b
            float z   = sigmoid_f(sl[10 + i]);     // slots 10..19 : iz+hz+b
            float hn  = sl[20 + i];                // slots 20..29 : h.whh_n + b_hh_n
            float inn = innB[i] + innW0[i] * xr0 + innW1[i] * xr1;
            float nn  = tanh_f(inn + r * hn);
            float ho  = Bv[i >> 1][i & 1];         // previous h_j (lane-local!)
            float h   = nn + z * (ho - nn);        // (1-z)*n + z*h
            hnew[i] = h;
            py0 += h * wl0[i];
            py1 += h * wl1[i];
        }
        // y = h_new @ w_l^T + b_l : combine the two lane halves
        float y0 = py0 + __shfl_xor(py0, 16, 32) + bl0;
        float y1 = py1 + __shfl_xor(py1, 16, 32) + bl1;

        if (g == 0 && inb) {
            size_t oidx = ((size_t)nb * step + (size_t)(step - 1 - ts)) * 2;
            *(float2*)(out + oidx) = make_float2(y0, y1);
        }

        // feed back: h_new slots are exactly next step's B operand elems
#pragma unroll
        for (int i = 0; i < 10; ++i) Bv[i >> 1][i & 1] = hnew[i];
        xr0 = y0; xr1 = y1;
        Bv[5][0] = g ? 1.0f : y0;
        Bv[5][1] = g ? 0.0f : y1;
    }
}

extern "C" void kernel_launch(void* const* d_in, const int* in_sizes, int n_in,
                              void* d_out, int out_size, void* d_ws, size_t ws_size,
                              hipStream_t stream) {
    const float* hidden = (const float*)d_in[0];
    const float* w_ih   = (const float*)d_in[1];
    const float* w_hh   = (const float*)d_in[2];
    const float* b_ih   = (const float*)d_in[3];
    const float* b_hh   = (const float*)d_in[4];
    const float* w_l    = (const float*)d_in[5];
    const float* b_l    = (const float*)d_in[6];
    const int*   step   = (const int*)d_in[7];
    float* out = (float*)d_out;

    const int B = in_sizes[0] / HDIM;              // hidden is (1, B, 20)
    const int waves  = (B + 15) / 16;              // 16 batch columns per wave
    const int wpb    = 8;                          // 256 threads = 8 waves
    const int blocks = (waves + wpb - 1) / wpb;

    gru_decoder_wmma<<<blocks, wpb * 32, 0, stream>>>(
        hidden, w_ih, w_hh, b_ih, b_hh, w_l, b_l, step, out, B);
}